// BehaviorFire_30296699306435
// MI455X (gfx1250) — compile-verified
//
#include <hip/hip_runtime.h>
#include <stdint.h>

#define BATCH 4
#define CCH   19
#define HH    1024
#define WW    1024
#define HWL   (HH*WW)
#define OFFC  5

#define TILE  32
#define HALO  3
#define PP    (TILE + 2*HALO)   // 38
#define NTHR  256
// 256 = RSTEP*PP + CSTEP  (for incremental r,c maintenance in plane loops)
#define RSTEP (NTHR / PP)       // 6
#define CSTEP (NTHR % PP)       // 28

// id codes: 0 empty, 1 wood, 2 plant, 3 gas, 4 dust, 5 ice, 6 fire, 7 lava,
// 8 water, 9 fish, 10 bird, 11 lemming, 12 kangaroo, 13 mole
__device__ __forceinline__ bool is_burnable_id(int i) {
    return (i==1)||(i==2)||(i==3)||(i==4)||(i==9)||(i==10)||(i==11)||(i==12)||(i==13);
}
__device__ __forceinline__ bool burn_cond_id(int i, float bc) {
    if (i == 4) return true;                  // dust always burns near fire
    if (i == 1 || i == 10) return bc < 0.05f; // wood, bird
    return bc < 0.2f;                         // plant, gas, other agents
}

// Wrap-around neighbor recompute straight from global memory. Cold path:
// only runs for cells on the image border (jnp.roll wraps, convs zero-pad).
// Keep it out of line so it isn't inlined 4x into the unrolled hot loop.
__device__ __noinline__ void bfdf_global(const float* __restrict__ idp,
                                         const float* __restrict__ bcp,
                                         int y, int x, float& bf, float& df) {
    bf = 0.f; df = 0.f;
    int i = (int)idp[(size_t)y*WW + x];
    if (!is_burnable_id(i)) return;
    bool hfn = false;
    for (int dy = -1; dy <= 1; ++dy) {
        for (int dx = -1; dx <= 1; ++dx) {
            int yy = y + dy, xx = x + dx;
            if (yy >= 0 && yy < HH && xx >= 0 && xx < WW) {
                int n = (int)idp[(size_t)yy*WW + xx];
                if (n == 6 || n == 7) hfn = true;
            }
        }
    }
    if (!hfn) return;
    float bc = bcp[(size_t)y*WW + x];
    if (burn_cond_id(i, bc)) {
        bf = 1.f;
        df = (i == 4) ? 1.f : 0.f;
    }
}

__global__ __launch_bounds__(NTHR)
void BehaviorFire_step(const float* __restrict__ world,
                       const float* __restrict__ rand_interact,
                       const float* __restrict__ rand_element,
                       float* __restrict__ out) {
    __shared__ float s_id [PP*PP];  // elem id (or -1 out of image)
    __shared__ float s_bc [PP*PP];  // burn chance
    __shared__ float s_fl [PP*PP];  // fire+lava pre; reused as (fwbn + lava)
    __shared__ float s_S1 [PP*PP];  // conv3(fire+lava)
    __shared__ float s_bp [PP*PP];  // burnables_post
    __shared__ float s_B1 [PP*PP];  // conv3(burnables_post)
    __shared__ float s_IFR[PP*PP];  // conv3(fwbn + lava)

    const int tx = threadIdx.x, ty = threadIdx.y;
    const int tid = ty*TILE + tx;
    const int bx = blockIdx.x, by = blockIdx.y, b = blockIdx.z;

    const float* win  = world + (size_t)b*CCH*HWL;
    float*       wout = out   + (size_t)b*CCH*HWL;
    const float* idp  = win;                              // channel-0 plane
    const float* bcp  = rand_interact + (size_t)b*HWL;
    const float* fep  = rand_element  + (size_t)b*HWL;

    const int gy0 = by*TILE, gx0 = bx*TILE;
    const int r0 = tid / PP, c0 = tid % PP;   // one div/mod for all plane passes

    // ---- Pass A: async-stage id + burn_chance halo planes into LDS (CDNA5 path)
    {
        uint64_t idbase = (uint64_t)idp;
        uint64_t bcbase = (uint64_t)bcp;
        for (int idx = tid, r = r0, c = c0; idx < PP*PP;
             idx += NTHR, r += RSTEP, c += CSTEP) {
            if (c >= PP) { c -= PP; ++r; }
            int gy = gy0 + r - HALO, gx = gx0 + c - HALO;
            if (gy >= 0 && gy < HH && gx >= 0 && gx < WW) {
                unsigned goff = (unsigned)((gy*WW + gx) * 4);
                unsigned lid  = (unsigned)(size_t)&s_id[idx];
                unsigned lbc  = (unsigned)(size_t)&s_bc[idx];
                asm volatile("global_load_async_to_lds_b32 %0, %1, %2"
                             :: "v"(lid), "v"(goff), "s"(idbase) : "memory");
                asm volatile("global_load_async_to_lds_b32 %0, %1, %2"
                             :: "v"(lbc), "v"(goff), "s"(bcbase) : "memory");
            } else {
                s_id[idx] = -1.f;   // sentinel: no type test matches
                s_bc[idx] = 1.f;
            }
        }
        asm volatile("s_wait_asynccnt 0" ::: "memory");
    }
    __syncthreads();

    // ---- Pass B: fire+lava indicator
    for (int idx = tid; idx < PP*PP; idx += NTHR) {
        int i = (int)s_id[idx];
        s_fl[idx] = (i == 6 || i == 7) ? 1.f : 0.f;
    }
    __syncthreads();

    // ---- Pass C: S1 = conv3(fire+lava), valid on rows/cols 1..PP-2
    for (int idx = tid, r = r0, c = c0; idx < PP*PP;
         idx += NTHR, r += RSTEP, c += CSTEP) {
        if (c >= PP) { c -= PP; ++r; }
        float s = 0.f;
        if (r >= 1 && r < PP-1 && c >= 1 && c < PP-1) {
            s = s_fl[idx-PP-1] + s_fl[idx-PP] + s_fl[idx-PP+1]
              + s_fl[idx-1]    + s_fl[idx]    + s_fl[idx+1]
              + s_fl[idx+PP-1] + s_fl[idx+PP] + s_fl[idx+PP+1];
        }
        s_S1[idx] = s;
    }
    __syncthreads();

    // ---- Pass D: burnables_post = burnable && !(cond && hfn)
    for (int idx = tid; idx < PP*PP; idx += NTHR) {
        float v = 0.f;
        int i = (int)s_id[idx];
        if (is_burnable_id(i)) {
            bool db = (s_S1[idx] > 0.f) && burn_cond_id(i, s_bc[idx]);
            v = db ? 0.f : 1.f;
        }
        s_bp[idx] = v;
    }
    __syncthreads();

    // ---- Pass E: B1 = conv3(burnables_post), valid on 2..PP-3
    for (int idx = tid, r = r0, c = c0; idx < PP*PP;
         idx += NTHR, r += RSTEP, c += CSTEP) {
        if (c >= PP) { c -= PP; ++r; }
        float s = 0.f;
        if (r >= 2 && r < PP-2 && c >= 2 && c < PP-2) {
            s = s_bp[idx-PP-1] + s_bp[idx-PP] + s_bp[idx-PP+1]
              + s_bp[idx-1]    + s_bp[idx]    + s_bp[idx+1]
              + s_bp[idx+PP-1] + s_bp[idx+PP] + s_bp[idx+PP+1];
        }
        s_B1[idx] = s;
    }
    __syncthreads();

    // ---- Pass F: fwbn + lava  (reuse s_fl; each lane touches only its own slot)
    for (int idx = tid, r = r0, c = c0; idx < PP*PP;
         idx += NTHR, r += RSTEP, c += CSTEP) {
        if (c >= PP) { c -= PP; ++r; }
        float v = 0.f;
        if (r >= 2 && r < PP-2 && c >= 2 && c < PP-2) {
            int i = (int)s_id[idx];
            v = (s_fl[idx] > 0.f ? s_B1[idx] : 0.f) + ((i == 7) ? 1.f : 0.f);
        }
        s_fl[idx] = v;
    }
    __syncthreads();

    // ---- Pass G: IFR = conv3(fwbn + lava), valid on the 32x32 center (3..PP-4)
    for (int idx = tid, r = r0, c = c0; idx < PP*PP;
         idx += NTHR, r += RSTEP, c += CSTEP) {
        if (c >= PP) { c -= PP; ++r; }
        float s = 0.f;
        if (r >= 3 && r < PP-3 && c >= 3 && c < PP-3) {
            s = s_fl[idx-PP-1] + s_fl[idx-PP] + s_fl[idx-PP+1]
              + s_fl[idx-1]    + s_fl[idx]    + s_fl[idx+1]
              + s_fl[idx+PP-1] + s_fl[idx+PP] + s_fl[idx+PP+1];
        }
        s_IFR[idx] = s;
    }
    __syncthreads();

    // ---- Final: per-cell decision + 19-channel streaming write
    auto bfdf_lds = [&](int p, float& bf, float& df) {
        int i = (int)s_id[p];
        bf = 0.f; df = 0.f;
        if (is_burnable_id(i) && s_S1[p] > 0.f && burn_cond_id(i, s_bc[p])) {
            bf = 1.f;
            df = (i == 4) ? 1.f : 0.f;
        }
    };

    #pragma unroll
    for (int k = 0; k < 4; ++k) {
        int ly = ty + 8*k;
        int gy = gy0 + ly, gx = gx0 + tx;
        int pr = ly + HALO, pc = tx + HALO;
        int pidx = pr*PP + pc;

        float idf = s_id[pidx];
        int ii = (int)idf;
        float bc = s_bc[pidx];
        bool hfn = s_S1[pidx] > 0.f;

        bool does_burn       = is_burnable_id(ii) && hfn && burn_cond_id(ii, bc);
        bool does_burn_ice   = (ii == 5) && (bc < 0.2f) && hfn;
        bool does_burn_empty = (ii == 0) && (s_IFR[pidx] > 0.f) && (bc < 0.3f);
        bool fire_post       = (ii == 6) || does_burn || does_burn_empty;
        float fe = fep[(size_t)gy*WW + gx];
        bool fire_to_empty   = fire_post && (fe < 0.4f) && (s_B1[pidx] == 0.f);

        bool repl = true; int newid = 0;
        if (fire_to_empty)                       newid = 0;  // empty
        else if (does_burn || does_burn_empty)   newid = 6;  // fire
        else if (does_burn_ice)                  newid = 8;  // water
        else                                     repl = false;

        size_t cell = (size_t)gy*WW + gx;
        float vely = 0.f, velx = 0.f;
        if (!repl) {
            float bfU,dfU,bfD,dfD,bfL,dfL,bfR,dfR;
            if (gy > 0)     bfdf_lds(pidx-PP, bfU, dfU); else bfdf_global(idp,bcp, HH-1, gx, bfU, dfU);
            if (gy < HH-1)  bfdf_lds(pidx+PP, bfD, dfD); else bfdf_global(idp,bcp, 0,    gx, bfD, dfD);
            if (gx > 0)     bfdf_lds(pidx-1,  bfL, dfL); else bfdf_global(idp,bcp, gy, WW-1, bfL, dfL);
            if (gx < WW-1)  bfdf_lds(pidx+1,  bfR, dfR); else bfdf_global(idp,bcp, gy, 0,    bfR, dfR);
            vely = win[(size_t)3*HWL + cell] + 8.f*(bfU - bfD) + 30.f*(dfU - dfD);
            velx = win[(size_t)4*HWL + cell] + 8.f*(bfL - bfR) + 30.f*(dfL - dfR);
        }

        #pragma unroll
        for (int ch = 0; ch < CCH; ++ch) {
            float v;
            if (repl)          v = (ch == 0) ? (float)newid : ((ch == OFFC + newid) ? 1.f : 0.f);
            else if (ch == 3)  v = vely;
            else if (ch == 4)  v = velx;
            else               v = win[(size_t)ch*HWL + cell];
            wout[(size_t)ch*HWL + cell] = v;
        }
    }
}

extern "C" void kernel_launch(void* const* d_in, const int* in_sizes, int n_in,
                              void* d_out, int out_size, void* d_ws, size_t ws_size,
                              hipStream_t stream) {
    (void)in_sizes; (void)n_in; (void)out_size; (void)d_ws; (void)ws_size;
    const float* world         = (const float*)d_in[0];
    // d_in[1] = rand_movement (unused by reference)
    const float* rand_interact = (const float*)d_in[2];
    const float* rand_element  = (const float*)d_in[3];
    // d_in[4..7] = kernel / fire_vec / water_vec / empty_vec (constants, hardcoded)
    float* out = (float*)d_out;

    dim3 grid(WW/TILE, HH/TILE, BATCH);
    dim3 block(TILE, 8, 1);
    hipLaunchKernelGGL(BehaviorFire_step, grid, block, 0, stream,
                       world, rand_interact, rand_element, out);
}